// PairList_72894184948381
// MI455X (gfx1250) — compile-verified
//
#include <hip/hip_runtime.h>
#include <hip/hip_bf16.h>
#include <math.h>

// PairList for N_ATOMS=8192, N_SUBSYSTEMS=64, CUTOFF=5.0.
// Outputs (concatenated flat, float): pair_i[P], pair_j[P], d_ij[P], r_ij[P*3], valid[P]
// P = n(n-1)/2 = 33,550,336  ->  7P floats ~ 939 MB written per call (store-BW bound).
//
// One wave32 per 16x16 tile of the upper-triangular pair space. Pairwise dot
// products via V_WMMA_F32_16X16X4_F32 (Gram-matrix trick):
//   d^2 = |pi|^2 + |pj|^2 - 2*pi.pj
// Rectangular grid (ti = blockIdx.y, tj = blockIdx.x*8+wave) with wave-uniform
// early-out replaces the triangular index decode. All pair indices fit int32;
// the flat pair index advances by (n - i - 2) per tile row. Non-temporal stores
// for the write-once output stream.

typedef __attribute__((ext_vector_type(2))) float v2f;
typedef __attribute__((ext_vector_type(8))) float v8f;

#define CUTOFF_F 5.0f
#define WAVES_PER_BLOCK 8

__global__ __launch_bounds__(256) void PairList_wmma_kernel(
    const float* __restrict__ pos,      // (n,3) row-major f32
    const int*   __restrict__ sub,      // (n,) subsystem ids
    float*       __restrict__ out,      // 7P floats
    int n, int T)
{
    const int lane = threadIdx.x & 31;
    const int wave = threadIdx.x >> 5;

    const int ti = blockIdx.y;
    const int tj = blockIdx.x * WAVES_PER_BLOCK + wave;
    if (tj < ti || tj >= T) return;     // wave-uniform: EXEC all-1s for WMMA

    const int i0 = ti * 16;
    const int j0 = tj * 16;

    const int half = lane >> 4;         // 0: lanes 0-15, 1: lanes 16-31
    const int l16  = lane & 15;

    // A tile (16x4 f32): rows i0..i0+15, K = {x,y,z,0}
    // lanes 0-15: v0=K0(x), v1=K1(y); lanes 16-31: v0=K2(z), v1=K3(pad)
    const int ar = min(i0 + l16, n - 1);
    v2f a;
    if (half == 0) { a.x = pos[3 * ar + 0]; a.y = pos[3 * ar + 1]; }
    else           { a.x = pos[3 * ar + 2]; a.y = 0.0f; }

    // B tile (4x16 f32): cols j0..j0+15, mirrored K packing on lane halves
    const int br = min(j0 + l16, n - 1);
    v2f b;
    if (half == 0) { b.x = pos[3 * br + 0]; b.y = pos[3 * br + 1]; }
    else           { b.x = pos[3 * br + 2]; b.y = 0.0f; }

    // Gram tile: acc[r] = dot(pos[i0 + 8*half + r], pos[j0 + l16])
    v8f acc = {};
    acc = __builtin_amdgcn_wmma_f32_16x16x4_f32(
        /*neg_a=*/false, a, /*neg_b=*/false, b,
        /*c_mod=*/(short)0, acc, /*reuse_a=*/false, /*reuse_b=*/false);

    // Per-lane column data (j fixed per lane)
    const int j  = j0 + l16;
    const int jc = min(j, n - 1);
    const float pjx = pos[3 * jc + 0];
    const float pjy = pos[3 * jc + 1];
    const float pjz = pos[3 * jc + 2];
    const float nj2 = pjx * pjx + pjy * pjy + pjz * pjz;
    const int   sj  = sub[jc];
    const float fj  = (float)j;

    const int P = (n / 2) * (n - 1);    // n even; fits int32
    float* __restrict__ outI = out;
    float* __restrict__ outJ = out + (size_t)P;
    float* __restrict__ outD = out + (size_t)2 * P;
    float* __restrict__ outR = out + (size_t)3 * P;
    float* __restrict__ outV = out + (size_t)6 * P;

    const int iBase = i0 + 8 * half;    // first row this lane-half covers
    // flat pair index for (iBase, j); i*(2n-i-1) is always even -> exact /2
    int p = (iBase * (2 * n - iBase - 1)) / 2 + (j - iBase - 1);
    const float fi0 = (float)iBase;

    const float* __restrict__ pi_ptr  = pos + 3 * iBase;  // row loads via imm offsets
    const int*   __restrict__ sub_i   = sub + iBase;

    const bool guarded = (ti == tj) | (i0 + 16 > n) | (j0 + 16 > n);

    if (!guarded) {
        // Fast path: every (iBase+r, j) is a strict upper-triangle pair.
#pragma unroll
        for (int r = 0; r < 8; ++r) {
            const int i = iBase + r;
            const float pix = pi_ptr[3 * r + 0];
            const float piy = pi_ptr[3 * r + 1];
            const float piz = pi_ptr[3 * r + 2];
            const float ni2 = pix * pix + piy * piy + piz * piz;

            float d2 = fmaf(-2.0f, acc[r], ni2 + nj2);
            d2 = fmaxf(d2, 0.0f);
            const float dij = __builtin_amdgcn_sqrtf(d2);   // single v_sqrt_f32

            const bool valid = (sub_i[r] == sj) && (dij < CUTOFF_F);
            const float m = valid ? 1.0f : 0.0f;

            __builtin_nontemporal_store(valid ? (fi0 + (float)r) : -1.0f, outI + p);
            __builtin_nontemporal_store(valid ? fj : -1.0f,               outJ + p);
            __builtin_nontemporal_store(dij * m,                          outD + p);
            __builtin_nontemporal_store((pjx - pix) * m,                  outR + 3 * p + 0);
            __builtin_nontemporal_store((pjy - piy) * m,                  outR + 3 * p + 1);
            __builtin_nontemporal_store((pjz - piz) * m,                  outR + 3 * p + 2);
            __builtin_nontemporal_store(m,                                outV + p);

            p += n - i - 2;             // p(i+1,j) = p(i,j) + (n - i - 2)
        }
    } else {
        // Diagonal / edge tiles: keep strict-upper-triangle + bounds guards.
#pragma unroll
        for (int r = 0; r < 8; ++r) {
            const int i = iBase + r;
            if (i < j && j < n && i < n) {
                const float pix = pi_ptr[3 * r + 0];
                const float piy = pi_ptr[3 * r + 1];
                const float piz = pi_ptr[3 * r + 2];
                const float ni2 = pix * pix + piy * piy + piz * piz;

                float d2 = fmaf(-2.0f, acc[r], ni2 + nj2);
                d2 = fmaxf(d2, 0.0f);
                const float dij = __builtin_amdgcn_sqrtf(d2);

                const bool valid = (sub_i[r] == sj) && (dij < CUTOFF_F);
                const float m = valid ? 1.0f : 0.0f;

                __builtin_nontemporal_store(valid ? (fi0 + (float)r) : -1.0f, outI + p);
                __builtin_nontemporal_store(valid ? fj : -1.0f,               outJ + p);
                __builtin_nontemporal_store(dij * m,                          outD + p);
                __builtin_nontemporal_store((pjx - pix) * m,                  outR + 3 * p + 0);
                __builtin_nontemporal_store((pjy - piy) * m,                  outR + 3 * p + 1);
                __builtin_nontemporal_store((pjz - piz) * m,                  outR + 3 * p + 2);
                __builtin_nontemporal_store(m,                                outV + p);
            }
            p += n - i - 2;
        }
    }
}

extern "C" void kernel_launch(void* const* d_in, const int* in_sizes, int n_in,
                              void* d_out, int out_size, void* d_ws, size_t ws_size,
                              hipStream_t stream) {
    const float* pos = (const float*)d_in[0];   // positions (n,3) f32
    const int*   sub = (const int*)d_in[1];     // atomic_subsystem_indices (n,)
    float* out = (float*)d_out;

    const int n = in_sizes[0] / 3;              // 8192
    const int T = (n + 15) / 16;                // 512 tiles per dim

    // Rectangular tile grid; waves with tj < ti exit immediately.
    dim3 grid((T + WAVES_PER_BLOCK - 1) / WAVES_PER_BLOCK, T);
    PairList_wmma_kernel<<<grid, 256, 0, stream>>>(pos, sub, out, n, T);
}